// MonoFlangerChorusModule_31568009626237
// MI455X (gfx1250) — compile-verified
//
#include <hip/hip_runtime.h>
#include <stdint.h>

// Flanger/chorus: D = 441 + 441 = 882 tap ring buffer, serial recurrence over N.
#define RING_D    882
#define MAX_LFO_F 441.0f
#define MAX_MIN_F 441.0f
#define TILE      1024
#define NTHREADS  256

// ---- CDNA5 async global->LDS staging -------------------------------------
// GV mode: global_load_async_to_lds_b32 VDST(lds byte addr), VADDR(64b), off
__device__ __forceinline__ void async_g2lds_b32(uint32_t lds_byte_addr, const float* g) {
    asm volatile("global_load_async_to_lds_b32 %0, %1, off"
                 :: "v"(lds_byte_addr), "v"((uint64_t)(uintptr_t)g)
                 : "memory");
}

#if defined(__has_builtin)
#  if __has_builtin(__builtin_amdgcn_s_wait_asynccnt)
#    define WAIT_ASYNC0() __builtin_amdgcn_s_wait_asynccnt(0)
#  endif
#endif
#ifndef WAIT_ASYNC0
#  define WAIT_ASYNC0() asm volatile("s_wait_asynccnt 0" ::: "memory")
#endif

// Truncate a generic pointer to a shared var down to its 32-bit LDS byte address.
__device__ __forceinline__ uint32_t lds_addr_of(const void* p) {
    return (uint32_t)(uintptr_t)p;
}

// Stage one tile: per-step packed {pi|ni<<16, frac_bits, x0_bits, x1_bits}.
// .x/.y written via normal DS stores; .z/.w DMA'd straight from global via
// async-to-LDS (tracked by ASYNCcnt, overlapped with the serial wave).
__device__ __forceinline__ void stage_tile(int4* __restrict__ st,
                                           const float* __restrict__ mb,
                                           const float* __restrict__ xb0,
                                           const float* __restrict__ xb1,
                                           int n0, int lim, int ntot,
                                           float dscale, float dmin,
                                           int lane, int stride)
{
    for (int i = lane; i < lim; i += stride) {
        const int n = n0 + i;
        const float mod = mb[n];
        // delay = (MAX_LFO*width)*mod + minw*MAX_MIN   (mul then add, like ref)
        const float dly = __fadd_rn(__fmul_rn(dscale, mod), dmin);
        const int   wi  = n % RING_D;
        // read_idx = (wi - delay + D) mod D, delay in [0, D)
        float t = __fadd_rn(__fsub_rn((float)wi, dly), (float)RING_D);
        if (t >= (float)RING_D) t -= (float)RING_D;
        const float pf = floorf(t);
        int pi = (int)pf;
        int ni = pi + 1; if (ni >= RING_D) ni = 0;
        st[i].x = pi | (ni << 16);
        st[i].y = __float_as_int(t - pf);

        // async DMA of both channels' samples into the packed element
        async_g2lds_b32(lds_addr_of(&st[i].z), xb0 + n);
        async_g2lds_b32(lds_addr_of(&st[i].w), xb1 + n);

        // pull the tile-after-next's mod signal toward L2 (global_prefetch_b8)
        if (n + 2 * TILE < ntot) __builtin_prefetch(mb + n + 2 * TILE, 0, 1);
    }
}

__global__ __launch_bounds__(NTHREADS, 1)
void flanger_kernel(const float* __restrict__ x,        // [B,2,N]
                    const float* __restrict__ mod_sig,  // [B,N]
                    const float* __restrict__ feedback, // [B]
                    const float* __restrict__ minw,     // [B]
                    const float* __restrict__ width,    // [B]
                    const float* __restrict__ depth,    // [B]
                    const float* __restrict__ mix,      // [B]
                    float* __restrict__ out,            // [B,2,N]
                    int ntot)
{
    __shared__ float s_buf[2][RING_D];   // delay rings, ch0/ch1 (bank-offset 50)
    __shared__ int4  s_st[2][TILE];      // double-buffered packed stage

    const int b   = blockIdx.x;
    const int tid = threadIdx.x;

    // zero-init delay rings (reference starts from a zero buffer)
    for (int i = tid; i < 2 * RING_D; i += NTHREADS)
        (&s_buf[0][0])[i] = 0.0f;

    // per-batch scalars (uniform -> scalar loads)
    const float fb     = feedback[b];
    const float dp     = depth[b];
    const float m      = mix[b];
    const float one_m  = __fsub_rn(1.0f, m);
    const float dscale = __fmul_rn(MAX_LFO_F, width[b]);
    const float dmin   = __fmul_rn(minw[b], MAX_MIN_F);

    const float* xb0 = x + (size_t)(2 * b + 0) * (size_t)ntot;
    const float* xb1 = x + (size_t)(2 * b + 1) * (size_t)ntot;
    const float* mb  = mod_sig + (size_t)b * (size_t)ntot;
    float*       ob  = out + (size_t)(2 * b) * (size_t)ntot;

    const int ntiles = (ntot + TILE - 1) / TILE;

    // ---- prologue: all 256 threads stage tile 0 into parity 0 ----
    {
        const int lim0 = (ntot < TILE) ? ntot : TILE;
        stage_tile(s_st[0], mb, xb0, xb1, 0, lim0, ntot, dscale, dmin,
                   tid, NTHREADS);
    }
    WAIT_ASYNC0();
    __syncthreads();

    // ---- pipelined main loop: wave0 lanes 0/1 run the serial recurrence on
    //      tile t (parity t&1) while waves 1..7 async-stage tile t+1 ----
    for (int t = 0; t < ntiles; ++t) {
        const int par = t & 1;
        const int n0  = t * TILE;
        const int lim = (ntot - n0 < TILE) ? (ntot - n0) : TILE;

        if (tid >= 32) {
            if (t + 1 < ntiles) {
                const int n1   = n0 + TILE;
                const int lim1 = (ntot - n1 < TILE) ? (ntot - n1) : TILE;
                stage_tile(s_st[1 - par], mb, xb0, xb1, n1, lim1, ntot,
                           dscale, dmin, tid - 32, NTHREADS - 32);
            }
        } else if (tid < 2) {
            const int c = tid;
            float* __restrict__ ring = s_buf[c];
            const int4* __restrict__ st = s_st[par];
            float* __restrict__ oc = ob + (size_t)c * (size_t)ntot + n0;
            int wi = n0 % RING_D;
            #pragma unroll 4
            for (int i = 0; i < lim; ++i) {
                const int4  pk = st[i];              // one ds_load_b128
                const int   pi = pk.x & 0xFFFF;
                const int   ni = pk.x >> 16;
                const float fr = __int_as_float(pk.y);
                const float xv = __int_as_float(c ? pk.w : pk.z);
                const float pv = ring[pi];
                const float nv = ring[ni];
                // interp = fr*nv + (1-fr)*pv   (match ref op order, no contraction)
                const float interp = __fadd_rn(__fmul_rn(fr, nv),
                                               __fmul_rn(__fsub_rn(1.0f, fr), pv));
                ring[wi] = __fadd_rn(xv, __fmul_rn(fb, interp));   // write AFTER reads
                const float outv = __fadd_rn(xv, __fmul_rn(dp, interp));
                float y = __fadd_rn(__fmul_rn(one_m, xv), __fmul_rn(m, outv));
                y = fminf(1.0f, fmaxf(-1.0f, y));
                oc[i] = y;
                ++wi; if (wi == RING_D) wi = 0;
            }
        }
        WAIT_ASYNC0();      // no-op for the serial wave (its ASYNCcnt is 0)
        __syncthreads();
    }
}

extern "C" void kernel_launch(void* const* d_in, const int* in_sizes, int n_in,
                              void* d_out, int out_size, void* d_ws, size_t ws_size,
                              hipStream_t stream) {
    const float* x        = (const float*)d_in[0];
    const float* mod_sig  = (const float*)d_in[1];
    const float* feedback = (const float*)d_in[2];
    const float* minw     = (const float*)d_in[3];
    const float* width    = (const float*)d_in[4];
    const float* depth    = (const float*)d_in[5];
    const float* mix      = (const float*)d_in[6];
    float* out = (float*)d_out;

    const int B = in_sizes[2];            // feedback has one element per batch
    const int N = in_sizes[1] / B;        // mod_sig is [B, N]

    flanger_kernel<<<dim3(B), dim3(NTHREADS), 0, stream>>>(
        x, mod_sig, feedback, minw, width, depth, mix, out, N);
}